// NoisyTopkRouter_10290741641430
// MI455X (gfx1250) — compile-verified
//
#include <hip/hip_runtime.h>
#include <hip/hip_bf16.h>
#include <math.h>

// ---------------- problem constants (match reference) ----------------
#define D_EMBED   2048
#define N_EXP     8
#define TOP_K     2
#define N_TOKENS  (4 * 4096)          // B*S = 16384

// ---------------- tiling ----------------
#define TILE_M        16              // tokens per wave (one WMMA tile)
#define WAVES_PER_WG  8               // 256 threads
#define WG_THREADS    (WAVES_PER_WG * 32)
#define N_TILES       (N_TOKENS / TILE_M)            // 1024
#define N_BLOCKS      (N_TILES / WAVES_PER_WG)       // 128

// LDS weight row stride: 2048 + 4 pad -> (2052 mod 64) == 4, so the 32 lanes'
// ds_load_b64 of a B-fragment touch all 64 banks exactly once (conflict-free).
#define WPAD 2052
#define TPAD 17   // epilogue tile row stride (17 mod 64 coprime-ish -> conflict-free)

typedef __attribute__((ext_vector_type(2))) float v2f;
typedef __attribute__((ext_vector_type(4))) float v4f;
typedef __attribute__((ext_vector_type(8))) float v8f;

// ---------------------------------------------------------------------------
// Kernel 0: zero the 16-float reduction workspace (prob sums [0..7], count
// sums [8..15]).
// ---------------------------------------------------------------------------
__global__ void router_zero_ws(float* __restrict__ red) {
    if (threadIdx.x < 16) red[threadIdx.x] = 0.0f;
}

// ---------------------------------------------------------------------------
// Kernel 1: fused routing GEMM (route + noise in one 16x16 WMMA tile) plus
// per-token epilogue (softplus noise, top-2, sparse softmax) and per-expert
// atomic accumulation for the load-balance loss.
// ---------------------------------------------------------------------------
__global__ __launch_bounds__(WG_THREADS) void router_main(
    const float* __restrict__ mh,        // [N_TOKENS, D]
    const float* __restrict__ noise,     // [N_TOKENS, E]
    const float* __restrict__ Wr,        // [E, D]
    const float* __restrict__ br,        // [E]
    const float* __restrict__ Wn,        // [E, D]
    const float* __restrict__ bn,        // [E]
    float* __restrict__ probs_out,       // [N_TOKENS, E]
    float* __restrict__ idx_out,         // [N_TOKENS, 2] (stored as float)
    float* __restrict__ red)             // [16] reduction ws
{
    // Combined weight matrix in LDS: rows 0..7 = W_route, rows 8..15 = W_noise.
    __shared__ float lds_w[16 * WPAD];                       // ~131 KB
    __shared__ float lds_t[WAVES_PER_WG * TILE_M * TPAD];    // ~8.5 KB

    const int tid = threadIdx.x;

    // ---- cooperative weight stage: 16 x 2048 f32, b128 loads + b128 LDS stores
    for (int i = tid; i < 16 * (D_EMBED / 4); i += WG_THREADS) {
        const int row = i >> 9;            // i / 512
        const int c4  = (i & 511) << 2;    // (i % 512) * 4
        const float* src = (row < 8 ? Wr + row * D_EMBED
                                    : Wn + (row - 8) * D_EMBED) + c4;
        v4f v = *(const v4f*)src;
        *(v4f*)&lds_w[row * WPAD + c4] = v;
    }
    __syncthreads();

    const int wave = tid >> 5;
    const int lane = tid & 31;
    const int row  = lane & 15;   // A: token-in-tile  | B: output column (expert)
    const int hi   = lane >> 4;   // 0: K={k,k+1}, 1: K={k+2,k+3}  (ISA fp32 layout)

    const int tile = blockIdx.x * WAVES_PER_WG + wave;
    const int tok0 = tile * TILE_M;

    const float* aptr = mh + (size_t)(tok0 + row) * D_EMBED + 2 * hi;
    const float* bptr = &lds_w[row * WPAD + 2 * hi];

    // ---- main K loop: D = A*B + C, A=16x4 activations, B=4x16 weights
    v8f acc = {};
    #pragma unroll 8
    for (int k = 0; k < D_EMBED; k += 4) {
        v2f a = *(const v2f*)(aptr + k);   // global_load_b64
        v2f b = *(const v2f*)(bptr + k);   // ds_load_b64, bank-conflict free
        acc = __builtin_amdgcn_wmma_f32_16x16x4_f32(
            /*neg_a=*/false, a, /*neg_b=*/false, b,
            /*c_mod=*/(short)0, acc, /*reuse_a=*/false, /*reuse_b=*/false);
    }

    // ---- spill C tile to LDS so each token's 16 logits are row-contiguous.
    // C layout: VGPR v -> M = v + 8*hi, N = row.
    float* tptr = &lds_t[wave * TILE_M * TPAD];
    #pragma unroll
    for (int v = 0; v < 8; ++v)
        tptr[(v + 8 * hi) * TPAD + row] = acc[v];
    asm volatile("s_wait_dscnt 0" ::: "memory");   // wave-local LDS RAW fence

    // ---- per-token epilogue: lanes 0..15 each own one token of the tile
    if (lane < 16) {
        const int m     = lane;
        const int token = tok0 + m;

        float noisy[N_EXP];
        const float* np = noise + (size_t)token * N_EXP;
        #pragma unroll
        for (int e = 0; e < N_EXP; ++e) {
            const float lg = tptr[m * TPAD + e] + br[e];            // route logit
            const float nl = tptr[m * TPAD + 8 + e] + bn[e];        // noise logit
            const float sp = (nl > 20.0f) ? nl : log1pf(expf(nl));  // softplus
            noisy[e] = lg + np[e] * sp;
        }

        // top-2, JAX tie-break = lowest index wins on equality (strict >)
        int i1 = 0; float v1 = noisy[0];
        #pragma unroll
        for (int e = 1; e < N_EXP; ++e)
            if (noisy[e] > v1) { v1 = noisy[e]; i1 = e; }
        int i2 = (i1 == 0) ? 1 : 0; float v2 = noisy[i2];
        #pragma unroll
        for (int e = 0; e < N_EXP; ++e)
            if (e != i1 && e != i2 && noisy[e] > v2) { v2 = noisy[e]; i2 = e; }

        // sparse softmax over the two kept logits (others are -inf -> 0)
        const float p2 = expf(v2 - v1);
        const float s  = 1.0f + p2;
        const float q1 = 1.0f / s;
        const float q2 = p2 / s;

        float pr[N_EXP];
        #pragma unroll
        for (int e = 0; e < N_EXP; ++e)
            pr[e] = (e == i1) ? q1 : ((e == i2) ? q2 : 0.0f);

        float* po = probs_out + (size_t)token * N_EXP;   // 32B aligned
        *(v4f*)(po)     = *(v4f*)&pr[0];                 // global_store_b128
        *(v4f*)(po + 4) = *(v4f*)&pr[4];

        idx_out[token * 2 + 0] = (float)i1;
        idx_out[token * 2 + 1] = (float)i2;

        // load-balance loss partials
        atomicAdd(&red[i1], q1);
        atomicAdd(&red[i2], q2);
        atomicAdd(&red[8 + i1], 1.0f);
        atomicAdd(&red[8 + i2], 1.0f);
    }
}

// ---------------------------------------------------------------------------
// Kernel 2: finalize loss = E * sum_e mean_prob[e] * mean_count[e]
// ---------------------------------------------------------------------------
__global__ void router_finalize(const float* __restrict__ red,
                                float* __restrict__ loss_out) {
    if (threadIdx.x == 0) {
        const float inv = 1.0f / (float)N_TOKENS;
        float sum = 0.0f;
        #pragma unroll
        for (int e = 0; e < N_EXP; ++e)
            sum += (red[e] * inv) * (red[8 + e] * inv);
        loss_out[0] = (float)N_EXP * sum;
    }
}

// ---------------------------------------------------------------------------
extern "C" void kernel_launch(void* const* d_in, const int* in_sizes, int n_in,
                              void* d_out, int out_size, void* d_ws, size_t ws_size,
                              hipStream_t stream) {
    const float* mh    = (const float*)d_in[0];   // [4,4096,2048]
    const float* noise = (const float*)d_in[1];   // [4,4096,8]
    const float* Wr    = (const float*)d_in[2];   // [8,2048]
    const float* br    = (const float*)d_in[3];   // [8]
    const float* Wn    = (const float*)d_in[4];   // [8,2048]
    const float* bn    = (const float*)d_in[5];   // [8]

    float* out   = (float*)d_out;
    float* probs = out;                                   // 131072 f32
    float* idx   = out + (size_t)N_TOKENS * N_EXP;        // 32768
    float* loss  = idx + (size_t)N_TOKENS * TOP_K;        // 1
    float* red   = (float*)d_ws;                          // 16 f32 scratch

    router_zero_ws<<<1, 32, 0, stream>>>(red);
    router_main<<<N_BLOCKS, WG_THREADS, 0, stream>>>(mh, noise, Wr, br, Wn, bn,
                                                     probs, idx, red);
    router_finalize<<<1, 32, 0, stream>>>(red, loss);
}